// ADMM_Net_20435454394561
// MI455X (gfx1250) — compile-verified
//
#include <hip/hip_runtime.h>
#include <float.h>

// ---------------------------------------------------------------------------
// ADMM deconvolution (diffuser-cam style) for MI455X / gfx1250.
// FFT2 is implemented as batched complex GEMMs against DFT matrices using
// V_WMMA_F32_16X16X4_F32 (full fp32 tensor op). DFT matrices (~18MB) stay
// resident in the 192MB L2; A and B tiles are staged through LDS so the inner
// loop is ds_load_b64 + v_wmma only.
// ---------------------------------------------------------------------------

constexpr int NPL  = 12;            // B*C planes
constexpr int D0   = 270, D1 = 480;
constexpr int P0   = 135, P1 = 240;
constexpr int N0   = 540, N1 = 960; // padded FFT grid
constexpr int M0P  = 544;           // 540 padded to multiple of 32
constexpr size_t PLANE = (size_t)M0P * N1;  // 522240 floats per plane
constexpr int ITERS = 5;

typedef __attribute__((ext_vector_type(2))) float v2f;
typedef __attribute__((ext_vector_type(8))) float v8f;

__device__ __forceinline__ v8f wmma4(v2f a, v2f b, v8f c) {
  // D = A(16x4,f32) * B(4x16,f32) + C(16x16,f32); NEG/OPSEL zero (only CNeg legal for F32)
  return __builtin_amdgcn_wmma_f32_16x16x4_f32(false, a, false, b, (short)0, c, false, false);
}

// ---------------------------------------------------------------------------
// Batched complex GEMM: C[p] = A[p] x B[p]  (A/B may broadcast with stride 0).
// A is MxK row-major (row stride K); B is KxN (row stride N).
// Block = 128 threads (4 waves), tile = 32(M) x 64(N), KB = 32.
//   - A slab (32xKB, re+im) staged in LDS row-major
//   - B slab (KBx64, re+im) staged in LDS TRANSPOSED so a B fragment is two
//     consecutive floats -> single ds_load_b64 with immediate offset
//   - each wave owns 2 M-tiles x 1 N-tile: 8 wmma per k-step (complex A)
// AREAL: A imag identically zero (forward FFT of a real field)
// STOREI: store the imaginary output plane
// ---------------------------------------------------------------------------
template<bool AREAL, bool STOREI>
__global__ __launch_bounds__(128)
void k_cgemm(const float* __restrict__ Ar, const float* __restrict__ Ai, size_t sA,
             const float* __restrict__ Br, const float* __restrict__ Bi, size_t sB,
             float* __restrict__ Cr, float* __restrict__ Ci, size_t sC,
             int M, int N, int K)
{
  __shared__ float lAr[32][34];
  __shared__ float lAi[32][34];
  __shared__ float lBr[64][34];
  __shared__ float lBi[64][34];

  const int p = blockIdx.z;
  const float* ar  = Ar + (size_t)p * sA;
  const float* ai  = Ai + (AREAL ? 0 : (size_t)p * sA);
  const float* br  = Br + (size_t)p * sB;
  const float* bim = Bi + (size_t)p * sB;
  float* crp = Cr + (size_t)p * sC;
  float* cip = Ci + (STOREI ? (size_t)p * sC : 0);

  const int lane  = threadIdx.x & 31;
  const int wave  = threadIdx.x >> 5;
  const int m0    = blockIdx.x * 32;
  const int n0blk = blockIdx.y * 64;
  const int hf    = lane >> 4;              // 0: lanes 0-15, 1: lanes 16-31
  const int ml    = lane & 15;              // A-frag row within M-tile
  const int nl    = wave * 16 + ml;         // column within block N-span

  v8f accR0 = {0.f,0.f,0.f,0.f,0.f,0.f,0.f,0.f};
  v8f accI0 = accR0, accR1 = accR0, accI1 = accR0;

  for (int kb = 0; kb < K; kb += 32) {
    // ---- stage A slab: 32 rows x 32 k (coalesced along k) ----
    for (int t = threadIdx.x; t < 32 * 32; t += 128) {
      int r = t >> 5, c = t & 31;
      size_t g = (size_t)(m0 + r) * K + (size_t)(kb + c);
      lAr[r][c] = ar[g];
      if constexpr (!AREAL) lAi[r][c] = ai[g];
    }
    // ---- stage B slab transposed: 32 k x 64 n (coalesced along n) ----
    for (int t = threadIdx.x; t < 32 * 64; t += 128) {
      int kk = t >> 6, n = t & 63;
      size_t g = (size_t)(kb + kk) * N + (size_t)(n0blk + n);
      lBr[n][kk] = br[g];
      lBi[n][kk] = bim[g];
    }
    __syncthreads();

    if (kb + 32 < K) {
      // hint next B k-slab toward L2 (global_prefetch_b8)
      __builtin_prefetch(&br[(size_t)(kb + 32 + (threadIdx.x >> 6)) * N + n0blk + (threadIdx.x & 63)], 0, 1);
    }

#pragma unroll
    for (int kk = 0; kk < 32; kk += 4) {
      const int k0 = kk + 2 * hf;           // ISA A-frag: lanes 16-31 hold K=2,3
      v2f bfr = { lBr[nl][k0], lBr[nl][k0 + 1] };
      v2f bfi = { lBi[nl][k0], lBi[nl][k0 + 1] };
      v2f ar0 = { lAr[ml][k0],      lAr[ml][k0 + 1]      };
      v2f ar1 = { lAr[16 + ml][k0], lAr[16 + ml][k0 + 1] };

      accR0 = wmma4(ar0, bfr, accR0);       // + Ar*Br
      accI0 = wmma4(ar0, bfi, accI0);       // + Ar*Bi
      accR1 = wmma4(ar1, bfr, accR1);
      accI1 = wmma4(ar1, bfi, accI1);

      if constexpr (!AREAL) {
        v2f ai0  = { lAi[ml][k0],      lAi[ml][k0 + 1]      };
        v2f ai1  = { lAi[16 + ml][k0], lAi[16 + ml][k0 + 1] };
        v2f nai0 = { -ai0.x, -ai0.y };
        v2f nai1 = { -ai1.x, -ai1.y };
        accR0 = wmma4(nai0, bfi, accR0);    // - Ai*Bi
        accI0 = wmma4(ai0,  bfr, accI0);    // + Ai*Br
        accR1 = wmma4(nai1, bfi, accR1);
        accI1 = wmma4(ai1,  bfr, accI1);
      }
    }
    __syncthreads();
  }

  // C/D layout: VGPR r -> row (tile_m0 + r + 8*hf), col = n
  const int nc    = n0blk + nl;
  const int mrow0 = m0 + 8 * hf;
  const int mrow1 = m0 + 16 + 8 * hf;
#pragma unroll
  for (int r = 0; r < 8; ++r) {
    size_t o0 = (size_t)(mrow0 + r) * N + nc;
    size_t o1 = (size_t)(mrow1 + r) * N + nc;
    crp[o0] = accR0[r];
    crp[o1] = accR1[r];
    if constexpr (STOREI) {
      cip[o0] = accI0[r];
      cip[o1] = accI1[r];
    }
  }
}

// ---------------------------------------------------------------------------
// Setup / elementwise kernels
// ---------------------------------------------------------------------------
__global__ void k_zero(float* p, size_t n) {
  size_t i = (size_t)blockIdx.x * blockDim.x + threadIdx.x;
  size_t st = (size_t)gridDim.x * blockDim.x;
  for (; i < n; i += st) p[i] = 0.0f;
}

// DFT matrix: wr + i*wi, entry [k][j] = scale * exp(sgn * 2*pi*i * k*j / n)
__global__ void k_genw(float* wr, float* wi, int n, int ld, float sgn, float scale) {
  size_t idx = (size_t)blockIdx.x * blockDim.x + threadIdx.x;
  size_t tot = (size_t)ld * ld;
  size_t st = (size_t)gridDim.x * blockDim.x;
  for (; idx < tot; idx += st) {
    int k = (int)(idx / ld), j = (int)(idx % ld);
    float cr = 0.f, ci = 0.f;
    if (k < n && j < n) {
      int t = (int)(((long long)k * (long long)j) % n);
      float x = 2.0f * (float)t / (float)n;     // angle / pi
      float s, c;
      sincospif(x, &s, &c);
      cr = scale * c;
      ci = sgn * scale * s;
    }
    wr[idx] = cr;
    wi[idx] = ci;
  }
}

// Cty = zero-pad(y) onto the 544x960 grid (rows >= 540 forced to zero)
__global__ void k_build_cty(const float* __restrict__ y, float* __restrict__ cty) {
  size_t idx = (size_t)blockIdx.x * blockDim.x + threadIdx.x;
  size_t tot = (size_t)NPL * PLANE;
  size_t st = (size_t)gridDim.x * blockDim.x;
  for (; idx < tot; idx += st) {
    size_t q = idx % PLANE;
    int p = (int)(idx / PLANE);
    int i = (int)(q / N1), j = (int)(q % N1);
    float v = 0.f;
    if (i >= P0 && i < P0 + D0 && j >= P1 && j < P1 + D1)
      v = y[(size_t)p * D0 * D1 + (size_t)(i - P0) * D1 + (j - P1)];
    cty[idx] = v;
  }
}

// fftshift(pad(h)) on 544x960 grid
__global__ void k_build_psf(const float* __restrict__ h, float* __restrict__ o) {
  size_t idx = (size_t)blockIdx.x * blockDim.x + threadIdx.x;
  size_t st = (size_t)gridDim.x * blockDim.x;
  for (; idx < PLANE; idx += st) {
    int i = (int)(idx / N1), j = (int)(idx % N1);
    float v = 0.f;
    if (i < N0) {
      int si = (i + N0 / 2) % N0;
      int sj = (j + N1 / 2) % N1;
      if (si >= P0 && si < P0 + D0 && sj >= P1 && sj < P1 + D1)
        v = h[(size_t)(si - P0) * D1 + (sj - P1)];
    }
    o[idx] = v;
  }
}

// Laplacian stencil on the padded grid
__global__ void k_build_lapl(float* __restrict__ o) {
  size_t idx = (size_t)blockIdx.x * blockDim.x + threadIdx.x;
  size_t st = (size_t)gridDim.x * blockDim.x;
  for (; idx < PLANE; idx += st) {
    int i = (int)(idx / N1), j = (int)(idx % N1);
    float v = 0.f;
    if (i == 0 && j == 0) v = 4.f;
    else if ((i == 0 && j == 1) || (i == 1 && j == 0)) v = -1.f;
    else if (i == 0 && j == N1 - 1) v = -1.f;
    else if (i == N0 - 1 && j == 0) v = -1.f;
    o[idx] = v;
  }
}

__global__ void k_mag2(const float* r, const float* im, float* o) {        // |F|^2
  size_t idx = (size_t)blockIdx.x * blockDim.x + threadIdx.x;
  size_t st = (size_t)gridDim.x * blockDim.x;
  for (; idx < PLANE; idx += st) o[idx] = r[idx] * r[idx] + im[idx] * im[idx];
}
__global__ void k_absmag(const float* r, const float* im, float* o) {      // |F|
  size_t idx = (size_t)blockIdx.x * blockDim.x + threadIdx.x;
  size_t st = (size_t)gridDim.x * blockDim.x;
  for (; idx < PLANE; idx += st) o[idx] = sqrtf(r[idx] * r[idx] + im[idx] * im[idx]);
}

// F *= H (sgn=+1) or F *= conj(H) (sgn=-1); H broadcast over planes
__global__ void k_cmul(float* __restrict__ fr, float* __restrict__ fi,
                       const float* __restrict__ hr, const float* __restrict__ hi,
                       float sgn) {
  size_t idx = (size_t)blockIdx.x * blockDim.x + threadIdx.x;
  size_t tot = (size_t)NPL * PLANE;
  size_t st = (size_t)gridDim.x * blockDim.x;
  for (; idx < tot; idx += st) {
    size_t q = idx % PLANE;
    float a = fr[idx], b = fi[idx], c = hr[q], d = sgn * hi[q];
    fr[idx] = a * c - b * d;
    fi[idx] = a * d + b * c;
  }
}

// F *= 1/(m1*HtH + m2*LtL + m3)
__global__ void k_smult(float* __restrict__ fr, float* __restrict__ fi,
                        const float* __restrict__ hth, const float* __restrict__ ltl,
                        const float* mu1, const float* mu2, const float* mu3, int it) {
  float m1 = mu1[it], m2 = mu2[it], m3 = mu3[it];
  size_t idx = (size_t)blockIdx.x * blockDim.x + threadIdx.x;
  size_t tot = (size_t)NPL * PLANE;
  size_t st = (size_t)gridDim.x * blockDim.x;
  for (; idx < tot; idx += st) {
    size_t q = idx % PLANE;
    float inv = 1.0f / (m1 * hth[q] + m2 * ltl[q] + m3);
    fr[idx] *= inv;
    fi[idx] *= inv;
  }
}

// ---- ADMM elementwise steps (over the logical 540x960 region) -------------
__device__ __forceinline__ void ewdec(size_t idx, int& p, int& i, int& j, size_t& base) {
  j = (int)(idx % N1);
  size_t t = idx / N1;
  i = (int)(t % N0);
  p = (int)(t / N0);
  base = (size_t)p * PLANE + (size_t)i * N1 + j;
}

__global__ void k_uupdate(const float* __restrict__ s,
                          const float* __restrict__ a21, const float* __restrict__ a22,
                          float* __restrict__ u1, float* __restrict__ u2,
                          const float* mu2, const float* tauv, int it) {
  float m2 = mu2[it], tau = tauv[it];
  size_t tot = (size_t)NPL * N0 * N1;
  size_t idx = (size_t)blockIdx.x * blockDim.x + threadIdx.x;
  size_t st = (size_t)gridDim.x * blockDim.x;
  for (; idx < tot; idx += st) {
    int p, i, j; size_t base; ewdec(idx, p, i, j, base);
    float sc = s[base];
    float vv = 0.f, hh = 0.f;
    if (i < N0 - 1) vv = -(s[base + N1] - sc) + a21[base] / m2;
    if (j < N1 - 1) hh = -(s[base + 1]  - sc) + a22[base] / m2;
    float mag = sqrtf(vv * vv + hh * hh);
    float magt = fmaxf(mag - tau, 0.f);
    float mm = magt / (magt + tau + 1e-5f);
    if (i < N0 - 1) u1[base] = mm * vv;
    if (j < N1 - 1) u2[base] = mm * hh;
  }
}

__global__ void k_vupdate(const float* __restrict__ a1, const float* __restrict__ Hs,
                          const float* __restrict__ cty, float* __restrict__ v,
                          const float* mu1, int it) {
  float m1 = mu1[it];
  size_t tot = (size_t)NPL * N0 * N1;
  size_t idx = (size_t)blockIdx.x * blockDim.x + threadIdx.x;
  size_t st = (size_t)gridDim.x * blockDim.x;
  for (; idx < tot; idx += st) {
    int p, i, j; size_t base; ewdec(idx, p, i, j, base);
    float ctc = (i >= P0 && i < P0 + D0 && j >= P1 && j < P1 + D1) ? 1.f : 0.f;
    v[base] = (a1[base] + m1 * Hs[base] + cty[base]) / (ctc + m1);
  }
}

__global__ void k_wupdate(const float* __restrict__ a3, const float* __restrict__ s,
                          float* __restrict__ w, const float* mu3, int it) {
  float m3 = mu3[it];
  size_t tot = (size_t)NPL * N0 * N1;
  size_t idx = (size_t)blockIdx.x * blockDim.x + threadIdx.x;
  size_t st = (size_t)gridDim.x * blockDim.x;
  for (; idx < tot; idx += st) {
    int p, i, j; size_t base; ewdec(idx, p, i, j, base);
    w[base] = fmaxf(a3[base] / m3 + s[base], 0.f);
  }
}

__global__ void k_t0(const float* __restrict__ v, const float* __restrict__ a1,
                     float* __restrict__ t0, const float* mu1, int it) {
  float m1 = mu1[it];
  size_t tot = (size_t)NPL * N0 * N1;
  size_t idx = (size_t)blockIdx.x * blockDim.x + threadIdx.x;
  size_t st = (size_t)gridDim.x * blockDim.x;
  for (; idx < tot; idx += st) {
    int p, i, j; size_t base; ewdec(idx, p, i, j, base);
    t0[base] = v[base] - a1[base] / m1;
  }
}

__global__ void k_num(const float* __restrict__ w, const float* __restrict__ a3,
                      const float* __restrict__ htmp,
                      const float* __restrict__ u1, const float* __restrict__ u2,
                      const float* __restrict__ a21, const float* __restrict__ a22,
                      float* __restrict__ num,
                      const float* mu1, const float* mu2, const float* mu3, int it) {
  float m1 = mu1[it], m2 = mu2[it], m3 = mu3[it];
  size_t tot = (size_t)NPL * N0 * N1;
  size_t idx = (size_t)blockIdx.x * blockDim.x + threadIdx.x;
  size_t st = (size_t)gridDim.x * blockDim.x;
  for (; idx < tot; idx += st) {
    int p, i, j; size_t base; ewdec(idx, p, i, j, base);
    size_t prow = (size_t)p * PLANE;
    float ax;
    if (i == 0) {
      size_t b = prow + j;
      ax = u1[b] - a21[b] / m2;
    } else if (i < N0 - 1) {
      size_t b  = prow + (size_t)i * N1 + j;
      size_t bm = b - N1;
      ax = (u1[b] - a21[b] / m2) - (u1[bm] - a21[bm] / m2);
    } else {
      size_t b = prow + (size_t)(N0 - 2) * N1 + j;
      ax = -(u1[b] - a21[b] / m2);
    }
    float bx;
    if (j == 0) {
      size_t b = prow + (size_t)i * N1;
      bx = u2[b] - a22[b] / m2;
    } else if (j < N1 - 1) {
      size_t b  = prow + (size_t)i * N1 + j;
      size_t bm = b - 1;
      bx = (u2[b] - a22[b] / m2) - (u2[bm] - a22[bm] / m2);
    } else {
      size_t b = prow + (size_t)i * N1 + (N1 - 2);
      bx = -(u2[b] - a22[b] / m2);
    }
    num[base] = m3 * w[base] - a3[base] + m1 * htmp[base] + m2 * (ax + bx);
  }
}

__global__ void k_dual(float* __restrict__ a1, float* __restrict__ a3,
                       float* __restrict__ a21, float* __restrict__ a22,
                       const float* __restrict__ s, const float* __restrict__ Hs,
                       const float* __restrict__ v, const float* __restrict__ w,
                       const float* __restrict__ u1, const float* __restrict__ u2,
                       const float* mu1, const float* mu2, const float* mu3, int it) {
  float m1 = mu1[it], m2 = mu2[it], m3 = mu3[it];
  size_t tot = (size_t)NPL * N0 * N1;
  size_t idx = (size_t)blockIdx.x * blockDim.x + threadIdx.x;
  size_t st = (size_t)gridDim.x * blockDim.x;
  for (; idx < tot; idx += st) {
    int p, i, j; size_t base; ewdec(idx, p, i, j, base);
    a1[base] += m1 * (Hs[base] - v[base]);
    a3[base] += m3 * (s[base] - w[base]);
    if (i < N0 - 1) a21[base] += m2 * (-(s[base + N1] - s[base]) - u1[base]);
    if (j < N1 - 1) a22[base] += m2 * (-(s[base + 1]  - s[base]) - u2[base]);
  }
}

// per-batch max over the cropped region (one block per batch)
__global__ void k_maxred(const float* __restrict__ s, float* __restrict__ maxb) {
  __shared__ float red[256];
  int b = blockIdx.x;
  float m = -FLT_MAX;
  const int per = 3 * D0 * D1;
  for (int t = threadIdx.x; t < per; t += 256) {
    int c = t / (D0 * D1);
    int r = t % (D0 * D1);
    int i = r / D1, j = r % D1;
    m = fmaxf(m, s[(size_t)(b * 3 + c) * PLANE + (size_t)(i + P0) * N1 + (j + P1)]);
  }
  red[threadIdx.x] = m;
  __syncthreads();
  for (int o = 128; o > 0; o >>= 1) {
    if (threadIdx.x < o) red[threadIdx.x] = fmaxf(red[threadIdx.x], red[threadIdx.x + o]);
    __syncthreads();
  }
  if (threadIdx.x == 0) maxb[b] = red[0];
}

__global__ void k_divout(const float* __restrict__ s, const float* __restrict__ maxb,
                         float* __restrict__ out) {
  size_t tot = (size_t)NPL * D0 * D1;
  size_t idx = (size_t)blockIdx.x * blockDim.x + threadIdx.x;
  size_t st = (size_t)gridDim.x * blockDim.x;
  for (; idx < tot; idx += st) {
    int j = (int)(idx % D1);
    size_t t = idx / D1;
    int i = (int)(t % D0);
    int p = (int)(t / D0);
    int b = p / 3;
    out[idx] = s[(size_t)p * PLANE + (size_t)(i + P0) * N1 + (j + P1)] / maxb[b];
  }
}

// ---------------------------------------------------------------------------
// Host orchestration
// ---------------------------------------------------------------------------
static inline void cgemm(hipStream_t st,
                         const float* Ar, const float* Ai, size_t sA,
                         const float* Br, const float* Bi, size_t sB,
                         float* Cr, float* Ci, size_t sC,
                         int M, int N, int K, int planes) {
  dim3 g(M / 32, N / 64, planes), b(128);
  // Ci must be a valid pointer even when unused by the template; reuse Cr.
  float* ci = Ci ? Ci : Cr;
  if (Ai) {
    if (Ci) k_cgemm<false, true ><<<g, b, 0, st>>>(Ar, Ai, sA, Br, Bi, sB, Cr, ci, sC, M, N, K);
    else    k_cgemm<false, false><<<g, b, 0, st>>>(Ar, Ai, sA, Br, Bi, sB, Cr, ci, sC, M, N, K);
  } else {
    const float* ai = Ar; // unused when AREAL
    if (Ci) k_cgemm<true,  true ><<<g, b, 0, st>>>(Ar, ai, sA, Br, Bi, sB, Cr, ci, sC, M, N, K);
    else    k_cgemm<true,  false><<<g, b, 0, st>>>(Ar, ai, sA, Br, Bi, sB, Cr, ci, sC, M, N, K);
  }
}

extern "C" void kernel_launch(void* const* d_in, const int* in_sizes, int n_in,
                              void* d_out, int out_size, void* d_ws, size_t ws_size,
                              hipStream_t stream) {
  const float* y    = (const float*)d_in[0];
  const float* h    = (const float*)d_in[1];
  const float* mu1  = (const float*)d_in[2];
  const float* mu2  = (const float*)d_in[3];
  const float* mu3  = (const float*)d_in[4];
  const float* tauv = (const float*)d_in[5];
  float* out = (float*)d_out;

  float* ws = (float*)d_ws;
  size_t off = 0;
  auto alloc = [&](size_t n) { float* p = ws + off; off += n; return p; };

  // DFT matrices (forward W = exp(-2pi i kj/n); inverse V = exp(+..)/n)
  float* W960r = alloc((size_t)960 * 960);
  float* W960i = alloc((size_t)960 * 960);
  float* V960r = alloc((size_t)960 * 960);
  float* V960i = alloc((size_t)960 * 960);
  float* W540r = alloc((size_t)M0P * M0P);
  float* W540i = alloc((size_t)M0P * M0P);
  float* V540r = alloc((size_t)M0P * M0P);
  float* V540i = alloc((size_t)M0P * M0P);

  // transfer functions (single plane)
  float* Hr  = alloc(PLANE);
  float* Hi  = alloc(PLANE);
  float* HtH = alloc(PLANE);
  float* LtL = alloc(PLANE);

  // persistent ADMM state -- zeroed every call (contiguous span)
  float* zbase = ws + off;
  float* s    = alloc(NPL * PLANE);
  float* a1   = alloc(NPL * PLANE);
  float* a3   = alloc(NPL * PLANE);
  float* Hskp = alloc(NPL * PLANE);
  float* a21  = alloc(NPL * PLANE);
  float* a22  = alloc(NPL * PLANE);
  float* num  = alloc(NPL * PLANE);   // also used as t0 (disjoint lifetimes)
  float* u1   = alloc(NPL * PLANE);
  float* u2   = alloc(NPL * PLANE);
  size_t zspan = (size_t)(ws + off - zbase);

  // fully-rewritten scratch
  float* Cty  = alloc(NPL * PLANE);
  float* v    = alloc(NPL * PLANE);
  float* w    = alloc(NPL * PLANE);
  float* Gr   = alloc(NPL * PLANE);
  float* Gi   = alloc(NPL * PLANE);
  float* Fr   = alloc(NPL * PLANE);
  float* Fi   = alloc(NPL * PLANE);
  float* htmp = alloc(NPL * PLANE);
  float* maxb = alloc(4);
  (void)ws_size; (void)in_sizes; (void)n_in; (void)out_size;

  const int TB = 256;
  const size_t nPlaneTot = (size_t)NPL * PLANE;
  const size_t nEW = (size_t)NPL * N0 * N1;
  dim3 bEW((unsigned)((nEW + TB - 1) / TB));
  dim3 bPl((unsigned)((nPlaneTot + TB - 1) / TB));
  dim3 b1P((unsigned)((PLANE + TB - 1) / TB));

  // --- init -----------------------------------------------------------------
  k_zero<<<dim3((unsigned)((zspan + TB - 1) / TB)), TB, 0, stream>>>(zbase, zspan);
  k_genw<<<dim3(3600), TB, 0, stream>>>(W960r, W960i, 960, 960, -1.f, 1.f);
  k_genw<<<dim3(3600), TB, 0, stream>>>(V960r, V960i, 960, 960, +1.f, 1.f / 960.f);
  k_genw<<<dim3(1156), TB, 0, stream>>>(W540r, W540i, 540, M0P, -1.f, 1.f);
  k_genw<<<dim3(1156), TB, 0, stream>>>(V540r, V540i, 540, M0P, +1.f, 1.f / 540.f);
  k_build_cty<<<bPl, TB, 0, stream>>>(y, Cty);

  // H = FFT2(fftshift(pad(h)));  single-plane FFT via GEMMs (scratch: Gr,Fr,Fi)
  k_build_psf<<<b1P, TB, 0, stream>>>(h, Gr);
  cgemm(stream, Gr, nullptr, 0, W960r, W960i, 0, Fr, Fi, 0, M0P, N1, N1, 1);
  cgemm(stream, W540r, W540i, 0, Fr, Fi, 0, Hr, Hi, 0, M0P, N1, M0P, 1);
  k_mag2<<<b1P, TB, 0, stream>>>(Hr, Hi, HtH);

  // LtL = |FFT2(laplacian stencil)|
  k_build_lapl<<<b1P, TB, 0, stream>>>(Gr);
  cgemm(stream, Gr, nullptr, 0, W960r, W960i, 0, Fr, Fi, 0, M0P, N1, N1, 1);
  cgemm(stream, W540r, W540i, 0, Fr, Fi, 0, Gi, htmp, 0, M0P, N1, M0P, 1);
  k_absmag<<<b1P, TB, 0, stream>>>(Gi, htmp, LtL);

  // --- ADMM iterations ------------------------------------------------------
  for (int it = 0; it < ITERS; ++it) {
    k_uupdate<<<bEW, TB, 0, stream>>>(s, a21, a22, u1, u2, mu2, tauv, it);
    k_vupdate<<<bEW, TB, 0, stream>>>(a1, Hskp, Cty, v, mu1, it);
    k_wupdate<<<bEW, TB, 0, stream>>>(a3, s, w, mu3, it);

    // htmp = Hadj(v - a1/m1) = ifft2(conj(H) * fft2(t0))
    k_t0<<<bEW, TB, 0, stream>>>(v, a1, num, mu1, it);  // num doubles as t0
    cgemm(stream, num, nullptr, PLANE, W960r, W960i, 0, Gr, Gi, PLANE, M0P, N1, N1, NPL);
    cgemm(stream, W540r, W540i, 0, Gr, Gi, PLANE, Fr, Fi, PLANE, M0P, N1, M0P, NPL);
    k_cmul<<<bPl, TB, 0, stream>>>(Fr, Fi, Hr, Hi, -1.f);
    cgemm(stream, Fr, Fi, PLANE, V960r, V960i, 0, Gr, Gi, PLANE, M0P, N1, N1, NPL);
    cgemm(stream, V540r, V540i, 0, Gr, Gi, PLANE, htmp, nullptr, PLANE, M0P, N1, M0P, NPL);

    // s-update: s = ifft2(Smult * fft2(num))
    k_num<<<bEW, TB, 0, stream>>>(w, a3, htmp, u1, u2, a21, a22, num, mu1, mu2, mu3, it);
    cgemm(stream, num, nullptr, PLANE, W960r, W960i, 0, Gr, Gi, PLANE, M0P, N1, N1, NPL);
    cgemm(stream, W540r, W540i, 0, Gr, Gi, PLANE, Fr, Fi, PLANE, M0P, N1, M0P, NPL);
    k_smult<<<bPl, TB, 0, stream>>>(Fr, Fi, HtH, LtL, mu1, mu2, mu3, it);
    cgemm(stream, Fr, Fi, PLANE, V960r, V960i, 0, Gr, Gi, PLANE, M0P, N1, N1, NPL);
    cgemm(stream, V540r, V540i, 0, Gr, Gi, PLANE, s, nullptr, PLANE, M0P, N1, M0P, NPL);

    // Hskp = Hfor(s)
    cgemm(stream, s, nullptr, PLANE, W960r, W960i, 0, Gr, Gi, PLANE, M0P, N1, N1, NPL);
    cgemm(stream, W540r, W540i, 0, Gr, Gi, PLANE, Fr, Fi, PLANE, M0P, N1, M0P, NPL);
    k_cmul<<<bPl, TB, 0, stream>>>(Fr, Fi, Hr, Hi, +1.f);
    cgemm(stream, Fr, Fi, PLANE, V960r, V960i, 0, Gr, Gi, PLANE, M0P, N1, N1, NPL);
    cgemm(stream, V540r, V540i, 0, Gr, Gi, PLANE, Hskp, nullptr, PLANE, M0P, N1, M0P, NPL);

    k_dual<<<bEW, TB, 0, stream>>>(a1, a3, a21, a22, s, Hskp, v, w, u1, u2,
                                   mu1, mu2, mu3, it);
  }

  // --- crop + per-batch normalization --------------------------------------
  k_maxred<<<dim3(4), dim3(256), 0, stream>>>(s, maxb);
  size_t nOut = (size_t)NPL * D0 * D1;
  k_divout<<<dim3((unsigned)((nOut + TB - 1) / TB)), TB, 0, stream>>>(s, maxb, out);
}